// RbfNet_22058952032725
// MI455X (gfx1250) — compile-verified
//
#include <hip/hip_runtime.h>
#include <hip/hip_bf16.h>

typedef __attribute__((ext_vector_type(2))) float v2f;
typedef __attribute__((ext_vector_type(8))) float v8f;

#define FEAT 32
#define NB 9   // SIZE*SIZE

// ---------------------------------------------------------------------------
// Stage 1: Xw[n][b][l] = sum_k X[n][k] * W[b][k][l]   (b=0..8, 32x32 GEMMs)
// One wave32 per 16-row tile of X. Uses V_WMMA_F32_16X16X4_F32, K chained 8x.
// Xw layout is node-major [N][9][32] so the edge kernel gathers 9 values with
// immediate offsets from one base address.
// ---------------------------------------------------------------------------
__global__ void __launch_bounds__(256)
xw_wmma_kernel(const float* __restrict__ X,   // [N,32]
               const float* __restrict__ W,   // [9,32,32]
               float* __restrict__ Xw,        // [N,9,32]
               int N)
{
    const int wave  = blockIdx.x * blockDim.y + threadIdx.y;
    const int tile0 = wave * 16;
    if (tile0 >= N) return;                    // wave-uniform: EXEC stays all-1

    const int lane  = threadIdx.x;             // 0..31 (wave32)
    const int mrow  = lane & 15;
    const int khalf = lane >> 4;               // 0 or 1
    const bool full = (tile0 + 16 <= N);       // wave-uniform

    int m = tile0 + mrow;
    if (m >= N) m = N - 1;                     // safety for partial tile loads

    // Preload all 8 A-fragments for this 16-row tile (reused 9 bases x 2 halves)
    // A 16x4 f32 layout: a[v] = A[m, kbase + v + 2*khalf]
    v2f a[8];
#pragma unroll
    for (int kb = 0; kb < 8; ++kb) {
        a[kb] = *(const v2f*)(X + (size_t)m * FEAT + kb * 4 + 2 * khalf);
    }

    const int n = lane & 15;
#pragma unroll
    for (int b = 0; b < NB; ++b) {
        const float* Wb = W + (size_t)b * FEAT * FEAT;   // [32,32] row-major (k,l)
#pragma unroll
        for (int nh = 0; nh < 2; ++nh) {
            const int ncol = nh * 16 + n;
            v8f c = {};
#pragma unroll
            for (int kb = 0; kb < 8; ++kb) {
                // B 4x16 f32 layout: bf[v] = B[kbase + v + 2*khalf, ncol]
                const int k0 = kb * 4 + 2 * khalf;
                v2f bf;
                bf.x = Wb[(size_t)k0 * FEAT + ncol];
                bf.y = Wb[(size_t)(k0 + 1) * FEAT + ncol];
                c = __builtin_amdgcn_wmma_f32_16x16x4_f32(
                        /*neg_a=*/false, a[kb],
                        /*neg_b=*/false, bf,
                        /*c_mod=*/(short)0, c,
                        /*reuse_a=*/false, /*reuse_b=*/false);
            }
            // D 16x16 f32 layout: c[v] = D[v + 8*khalf, ncol]
            // Xw[node][b][ncol], node = tile0 + v + 8*khalf
            float* dst = Xw + ((size_t)(tile0 + 8 * khalf) * NB + b) * FEAT + ncol;
            if (full) {
#pragma unroll
                for (int v = 0; v < 8; ++v)
                    dst[(size_t)v * NB * FEAT] = c[v];   // +v*1152B immediates
            } else {
#pragma unroll
                for (int v = 0; v < 8; ++v) {
                    const int mm = tile0 + v + 8 * khalf;
                    if (mm < N)
                        dst[(size_t)v * NB * FEAT] = c[v];
                }
            }
        }
    }
}

// ---------------------------------------------------------------------------
// Zero the accumulator
// ---------------------------------------------------------------------------
__global__ void zero_kernel(float* __restrict__ p, int n)
{
    int i = blockIdx.x * blockDim.x + threadIdx.x;
    if (i < n) p[i] = 0.0f;
}

// ---------------------------------------------------------------------------
// Stage 2: per-edge scatter.  acc[col] += sum_b basis[e,b] * Xw[row][b]
// One wave32 per edge; lane = feature channel. Xw gathers are L2-resident:
// 9 global_load_b32 at immediate offsets from one base (1152B per source row).
// ---------------------------------------------------------------------------
__global__ void __launch_bounds__(256)
edge_kernel(const float* __restrict__ positions,  // [Ntot,2]
            const float* __restrict__ outpos,     // [M,2]
            const float* __restrict__ support_p,  // [1]
            const int*   __restrict__ edges,      // row[E], col[E]
            const float* __restrict__ Xw,         // [Ntot,9,32]
            float* __restrict__ acc,              // [M,32]
            int E)
{
    const int e = blockIdx.x * blockDim.y + threadIdx.y;
    if (e >= E) return;
    const int lane = threadIdx.x;

    const int row = edges[e];
    const int col = edges[E + e];

    const float inv_s = 1.0f / support_p[0];
    const float u_raw = (outpos[col * 2 + 0] - positions[row * 2 + 0]) * inv_s;
    const float v_raw = (outpos[col * 2 + 1] - positions[row * 2 + 1]) * inv_s;
    const float u = fminf(fmaxf(u_raw, -1.0f), 1.0f);
    const float v = fminf(fmaxf(v_raw, -1.0f), 1.0f);

    // centers {-1,0,1}, gamma = 1: phi_i = exp(-(p - c_i)^2)
    float phiu[3], phiv[3];
    phiu[0] = __expf(-(u + 1.0f) * (u + 1.0f));
    phiu[1] = __expf(-u * u);
    phiu[2] = __expf(-(u - 1.0f) * (u - 1.0f));
    phiv[0] = __expf(-(v + 1.0f) * (v + 1.0f));
    phiv[1] = __expf(-v * v);
    phiv[2] = __expf(-(v - 1.0f) * (v - 1.0f));

    const float* xwr = Xw + (size_t)row * (NB * FEAT) + lane;

    float s = 0.0f;
#pragma unroll
    for (int i = 0; i < 3; ++i)
#pragma unroll
        for (int j = 0; j < 3; ++j) {
            const float bas = phiu[i] * phiv[j];
            s = fmaf(bas, xwr[(size_t)(i * 3 + j) * FEAT], s);
        }

    unsafeAtomicAdd(&acc[(size_t)col * FEAT + lane], s);
}

// ---------------------------------------------------------------------------
// Stage 3: ReLU (layers 0,1) + ghost-row expansion into the next layer input
// ---------------------------------------------------------------------------
__global__ void finalize_kernel(const float* __restrict__ acc,       // [M,32]
                                const int*   __restrict__ ghostIdx,  // [Ntot]
                                float* __restrict__ out,             // [Ntot,32]
                                int M, int Ntot, int act)
{
    const int idx = blockIdx.x * blockDim.x + threadIdx.x;
    if (idx >= Ntot * FEAT) return;
    const int nf = idx >> 5;
    const int l  = idx & 31;
    const int src = (nf < M) ? nf : ghostIdx[nf];
    float v = acc[(size_t)src * FEAT + l];
    if (act) v = fmaxf(v, 0.0f);
    out[idx] = v;
}

// ---------------------------------------------------------------------------
extern "C" void kernel_launch(void* const* d_in, const int* in_sizes, int n_in,
                              void* d_out, int out_size, void* d_ws, size_t ws_size,
                              hipStream_t stream)
{
    const float* positions = (const float*)d_in[0];
    const float* features  = (const float*)d_in[1];
    const float* outpos    = (const float*)d_in[2];
    const int*   ghostIdx  = (const int*)  d_in[3];
    const float* support   = (const float*)d_in[4];
    const int*   edges     = (const int*)  d_in[5];
    const float* Wptr[3]   = { (const float*)d_in[6],
                               (const float*)d_in[7],
                               (const float*)d_in[8] };

    const int Ntot = in_sizes[0] / 2;   // 60000
    const int M    = in_sizes[2] / 2;   // 54000
    const int E    = in_sizes[5] / 2;   // 1000000

    // Workspace layout (all f32):
    //   Xw  : Ntot * 9 * 32   (node-major)
    //   acc : M * 32
    //   ans : Ntot * 32
    float* Xw  = (float*)d_ws;
    float* acc = Xw + (size_t)Ntot * NB * FEAT;
    float* ans = acc + (size_t)M * FEAT;

    const int tiles = (Ntot + 15) / 16;
    const dim3 wblk(32, 8);                          // 8 waves / block
    const int  wgrd = (tiles + 7) / 8;
    const int  egrd = (E + 7) / 8;
    const int  accN = M * FEAT;
    const int  ansN = Ntot * FEAT;

    for (int layer = 0; layer < 3; ++layer) {
        const float* X = (layer == 0) ? features : ans;
        float* dst = (layer == 2) ? (float*)d_out : ans;

        xw_wmma_kernel<<<wgrd, wblk, 0, stream>>>(X, Wptr[layer], Xw, Ntot);
        zero_kernel<<<(accN + 255) / 256, 256, 0, stream>>>(acc, accN);
        edge_kernel<<<egrd, wblk, 0, stream>>>(positions, outpos, support,
                                               edges, Xw, acc, E);
        finalize_kernel<<<(ansN + 255) / 256, 256, 0, stream>>>(
            acc, ghostIdx, dst, M, Ntot, layer < 2 ? 1 : 0);
    }
}